// SkewedLossFunction_OneSide_44143673869005
// MI455X (gfx1250) — compile-verified
//
#include <hip/hip_runtime.h>
#include <hip/hip_bf16.h>

typedef __attribute__((ext_vector_type(2))) float v2f;
typedef __attribute__((ext_vector_type(8))) float v8f;

#define BLOCK_SIZE 256
#define GRID_SIZE  2048

// skewed one-sided loss per element:
// lam = t/50 - 1 ; gate = 1 iff sign(t-p)==sign(lam)!=0 ; loss = |t-p| * exp(gate*|lam|)
__device__ __forceinline__ float skew_elem(float p, float t) {
    float d   = t - p;
    float lam = __fmaf_rn(t, 0.02f, -1.0f);
    float ad  = __builtin_fabsf(d);
    float al  = __builtin_fabsf(lam);
    // same nonzero sign <=> d*lam > 0
    float mult = (d * lam > 0.0f) ? __expf(al) : 1.0f;
    return ad * mult;
}

__global__ void __launch_bounds__(BLOCK_SIZE)
skew_partial_kernel(const float4* __restrict__ p4,
                    const float4* __restrict__ t4,
                    float* __restrict__ partials,
                    int n4, int n_total) {
    int tid    = blockIdx.x * BLOCK_SIZE + threadIdx.x;
    int stride = gridDim.x * BLOCK_SIZE;

    float acc = 0.0f;
    // grid-stride over float4s -> global_load_b128 streams, bandwidth-bound
    for (int i = tid; i < n4; i += stride) {
        float4 p = p4[i];
        float4 t = t4[i];
        acc += skew_elem(p.x, t.x);
        acc += skew_elem(p.y, t.y);
        acc += skew_elem(p.z, t.z);
        acc += skew_elem(p.w, t.w);
    }
    // scalar tail (none for N = 2^23, but keep general)
    if (tid == 0) {
        const float* pf = (const float*)p4;
        const float* tf = (const float*)t4;
        for (int i = n4 * 4; i < n_total; ++i) acc += skew_elem(pf[i], tf[i]);
    }

    // wave32 tree reduction (deterministic order)
    #pragma unroll
    for (int off = 16; off > 0; off >>= 1)
        acc += __shfl_down(acc, off, 32);

    __shared__ float wsum[BLOCK_SIZE / 32];
    int lane = threadIdx.x & 31;
    int wid  = threadIdx.x >> 5;
    if (lane == 0) wsum[wid] = acc;
    __syncthreads();

    if (wid == 0) {
        float v = (lane < (BLOCK_SIZE / 32)) ? wsum[lane] : 0.0f;
        #pragma unroll
        for (int off = 4; off > 0; off >>= 1)
            v += __shfl_down(v, off, 32);
        if (lane == 0) partials[blockIdx.x] = v;
    }
}

// Final reduction: one wave32. Lane sums go into B of V_WMMA_F32_16X16X4_F32
// with A = ones: every row of D is the vector of B's column sums, so summing
// one row of D (c[0] over lanes 0..15) gives the exact f32 total.
__global__ void __launch_bounds__(32)
skew_final_kernel(const float* __restrict__ partials,
                  int n_partials,
                  float* __restrict__ out,
                  float inv_n) {
    int lane = threadIdx.x;  // 0..31, EXEC all-ones for WMMA
    float s = 0.0f;
    for (int i = lane; i < n_partials; i += 32)
        s += partials[i];

    v2f a; a.x = 1.0f; a.y = 1.0f;   // A (16x4) = all ones
    v2f b; b.x = s;    b.y = 0.0f;   // B (4x16): 32 lane sums + 32 zeros
    v8f c = {};
    // (neg_a, A, neg_b, B, c_mod, C, reuse_a, reuse_b)
    c = __builtin_amdgcn_wmma_f32_16x16x4_f32(
        false, a, false, b, (short)0, c, false, false);

    // c[0] lanes 0..15 hold the 16 column sums; reduce within the 16-lane half
    float col = c[0];
    col += __shfl_xor(col, 1, 32);
    col += __shfl_xor(col, 2, 32);
    col += __shfl_xor(col, 4, 32);
    col += __shfl_xor(col, 8, 32);

    if (lane == 0) out[0] = col * inv_n;
}

extern "C" void kernel_launch(void* const* d_in, const int* in_sizes, int n_in,
                              void* d_out, int out_size, void* d_ws, size_t ws_size,
                              hipStream_t stream) {
    const float* y_pred = (const float*)d_in[0];
    const float* y_true = (const float*)d_in[1];
    float* out = (float*)d_out;
    int n  = in_sizes[0];
    int n4 = n >> 2;

    float* partials = (float*)d_ws;  // GRID_SIZE floats (8 KB)

    skew_partial_kernel<<<GRID_SIZE, BLOCK_SIZE, 0, stream>>>(
        (const float4*)y_pred, (const float4*)y_true, partials, n4, n);

    float inv_n = 1.0f / (float)n;   // N = 2^23 -> exact
    skew_final_kernel<<<1, 32, 0, stream>>>(partials, GRID_SIZE, out, inv_n);
}